// SimplifiedGATModel_68281390072315
// MI455X (gfx1250) — compile-verified
//
#include <hip/hip_runtime.h>
#include <hip/hip_bf16.h>
#include <math.h>

typedef __attribute__((ext_vector_type(2))) float v2f;
typedef __attribute__((ext_vector_type(8))) float v8f;

#define NEG_SLOPE 0.2f

// ---------- helpers: monotone uint key for float atomic max ----------
__device__ __forceinline__ unsigned f2key(float f) {
    unsigned u = __float_as_uint(f);
    return (u & 0x80000000u) ? ~u : (u | 0x80000000u);
}
__device__ __forceinline__ float key2f(unsigned k) {
    return (k & 0x80000000u) ? __uint_as_float(k & 0x7FFFFFFFu)
                             : __uint_as_float(~k);
}

// ---------- generic zero ----------
__global__ void k_zero(float* p, long long n) {
    long long i = (long long)blockIdx.x * blockDim.x + threadIdx.x;
    long long stride = (long long)gridDim.x * blockDim.x;
    for (; i < n; i += stride) p[i] = 0.0f;
}

// ---------- self-loop attr = mean of incoming edge attrs ----------
__global__ void k_loop_acc(const int* __restrict__ dst, const float* __restrict__ ea,
                           float* __restrict__ sum, float* __restrict__ cnt, int E) {
    int e = blockIdx.x * blockDim.x + threadIdx.x;
    if (e >= E) return;
    int d = dst[e];
    atomicAdd(&sum[d * 3 + 0], ea[e * 3 + 0]);
    atomicAdd(&sum[d * 3 + 1], ea[e * 3 + 1]);
    atomicAdd(&sum[d * 3 + 2], ea[e * 3 + 2]);
    atomicAdd(&cnt[d], 1.0f);
}
__global__ void k_loop_fin(const float* __restrict__ sum, const float* __restrict__ cnt,
                           float* __restrict__ loop_attr, int N) {
    int n = blockIdx.x * blockDim.x + threadIdx.x;
    if (n >= N) return;
    float c = fmaxf(cnt[n], 1.0f);
    loop_attr[n * 3 + 0] = sum[n * 3 + 0] / c;
    loop_attr[n * 3 + 1] = sum[n * 3 + 1] / c;
    loop_attr[n * 3 + 2] = sum[n * 3 + 2] / c;
}

// ---------- we_att[j,h] = sum_c We[j, h*C+c] * att_e[h,c]  (3 x H, tiny) ----------
__global__ void k_we_att(const float* __restrict__ We, const float* __restrict__ att_e,
                         float* __restrict__ we_att, int H, int C) {
    int idx = threadIdx.x;
    if (idx >= 3 * H) return;
    int j = idx / H, h = idx % H;
    float s = 0.0f;
    for (int c = 0; c < C; ++c) s += We[j * (H * C) + h * C + c] * att_e[h * C + c];
    we_att[j * H + h] = s;
}

// ---------- fp32 WMMA GEMM: C[M,Nn] = A[M,K] @ B[K,Nn], row-major ----------
// One wave32 per 16x16 tile, V_WMMA_F32_16X16X4_F32, K stepped by 4.
__global__ void k_gemm_wmma_f32(const float* __restrict__ A, const float* __restrict__ B,
                                float* __restrict__ C, int M, int Nn, int K) {
    int wavesPerBlock = blockDim.x >> 5;
    long long wave = (long long)blockIdx.x * wavesPerBlock + (threadIdx.x >> 5);
    int lane = threadIdx.x & 31;
    int ntiles = Nn >> 4;
    int mtiles = (M + 15) >> 4;
    if (wave >= (long long)mtiles * ntiles) return;
    int mt = (int)(wave / ntiles), nt = (int)(wave % ntiles);
    int row0 = mt << 4, col0 = nt << 4;
    int l16 = lane & 15;   // m (for A/D) or n (for B/D)
    int g   = lane >> 4;   // half-wave group

    // A frag row (clamped so EXEC stays full; store is guarded instead)
    int arow = row0 + l16; if (arow >= M) arow = M - 1;
    const float* Arow = A + (long long)arow * K;

    v8f acc = {};
    for (int k0 = 0; k0 < K; k0 += 4) {
        v2f a, b;
        // A 16x4: lanes 0-15 hold K={k0,k0+1}, lanes 16-31 hold K={k0+2,k0+3}
        a.x = Arow[k0 + 2 * g + 0];
        a.y = Arow[k0 + 2 * g + 1];
        // B 4x16: reg r, group g -> K row = k0 + r + 2*g, N col = col0 + l16
        b.x = B[(long long)(k0 + 2 * g + 0) * Nn + col0 + l16];
        b.y = B[(long long)(k0 + 2 * g + 1) * Nn + col0 + l16];
        acc = __builtin_amdgcn_wmma_f32_16x16x4_f32(
            /*neg_a=*/false, a, /*neg_b=*/false, b,
            /*c_mod=*/(short)0, acc, /*reuse_a=*/false, /*reuse_b=*/false);
    }
    // D 16x16: VGPR v -> row v + 8*g, col = l16
#pragma unroll
    for (int v = 0; v < 8; ++v) {
        int row = row0 + v + 8 * g;
        if (row < M) C[(long long)row * Nn + col0 + l16] = acc[v];
    }
}

// ---------- per-node attention terms a_s, a_d ----------
__global__ void k_node_att(const float* __restrict__ xl, const float* __restrict__ att_s,
                           const float* __restrict__ att_d, float* __restrict__ a_s,
                           float* __restrict__ a_d, int N, int H, int C) {
    long long idx = (long long)blockIdx.x * blockDim.x + threadIdx.x;
    if (idx >= (long long)N * H) return;
    int n = (int)(idx / H), h = (int)(idx % H);
    const float* xr = xl + (long long)n * H * C + h * C;
    float ss = 0.0f, sd = 0.0f;
    for (int c = 0; c < C; ++c) {
        float v = xr[c];
        ss += v * att_s[h * C + c];
        sd += v * att_d[h * C + c];
    }
    a_s[idx] = ss;
    a_d[idx] = sd;
}

// ---------- edge pass 1: leakyrelu logit, store, atomic segment-max ----------
__global__ void k_edge_logit(const int* __restrict__ src, const int* __restrict__ dst,
                             const float* __restrict__ edge_attr,
                             const float* __restrict__ loop_attr,
                             const float* __restrict__ we_att,
                             const float* __restrict__ a_s, const float* __restrict__ a_d,
                             float* __restrict__ logits, unsigned* __restrict__ mkey,
                             int E, int N, int H) {
    long long idx = (long long)blockIdx.x * blockDim.x + threadIdx.x;
    long long total = (long long)(E + N) * H;
    if (idx >= total) return;
    int e = (int)(idx / H), h = (int)(idx % H);
    int s, d;
    const float* ea;
    if (e < E) { s = src[e]; d = dst[e]; ea = edge_attr + (long long)e * 3; }
    else       { s = d = e - E;          ea = loop_attr + (long long)(e - E) * 3; }
    float aev = ea[0] * we_att[0 * H + h] + ea[1] * we_att[1 * H + h] + ea[2] * we_att[2 * H + h];
    float z = a_s[(long long)s * H + h] + a_d[(long long)d * H + h] + aev;
    z = (z > 0.0f) ? z : NEG_SLOPE * z;
    logits[idx] = z;
    atomicMax(&mkey[(long long)d * H + h], f2key(z));
}

// ---------- edge pass 2: exp(logit - max), store, atomic segment-sum ----------
__global__ void k_edge_exp(const int* __restrict__ dst, float* __restrict__ logits,
                           const unsigned* __restrict__ mkey, float* __restrict__ denom,
                           int E, int N, int H) {
    long long idx = (long long)blockIdx.x * blockDim.x + threadIdx.x;
    long long total = (long long)(E + N) * H;
    if (idx >= total) return;
    int e = (int)(idx / H), h = (int)(idx % H);
    int d = (e < E) ? dst[e] : (e - E);
    float m = key2f(mkey[(long long)d * H + h]);
    float ex = __expf(logits[idx] - m);
    logits[idx] = ex;
    atomicAdd(&denom[(long long)d * H + h], ex);
}

// ---------- edge pass 3: out[dst] += alpha * xl[src]  (hc fastest -> coalesced) ----------
__global__ void k_edge_scatter(const int* __restrict__ src, const int* __restrict__ dst,
                               const float* __restrict__ ex, const float* __restrict__ denom,
                               const float* __restrict__ xl, float* __restrict__ out,
                               int E, int N, int H, int C) {
    int HC = H * C;
    long long idx = (long long)blockIdx.x * blockDim.x + threadIdx.x;
    long long total = (long long)(E + N) * HC;
    long long stride = (long long)gridDim.x * blockDim.x;
    for (; idx < total; idx += stride) {
        int hc = (int)(idx % HC);
        long long e = idx / HC;
        int h = hc / C;
        int s, d;
        if (e < E) { s = src[e]; d = dst[e]; }
        else       { s = d = (int)(e - E); }
        float alpha = ex[e * H + h] / (denom[(long long)d * H + h] + 1e-16f);
        atomicAdd(&out[(long long)d * HC + hc], alpha * xl[(long long)s * HC + hc]);
    }
}

// ---------- epilogue: out += bias, optional ELU ----------
__global__ void k_bias_elu(float* __restrict__ h, const float* __restrict__ b,
                           long long n, int HC, int do_elu) {
    long long i = (long long)blockIdx.x * blockDim.x + threadIdx.x;
    long long stride = (long long)gridDim.x * blockDim.x;
    for (; i < n; i += stride) {
        float v = h[i] + b[(int)(i % HC)];
        if (do_elu) v = (v > 0.0f) ? v : expm1f(v);
        h[i] = v;
    }
}

static inline unsigned ugrid(long long work, int tpb) {
    long long b = (work + tpb - 1) / tpb;
    if (b > (1LL << 20)) b = (1LL << 20);
    if (b < 1) b = 1;
    return (unsigned)b;
}

extern "C" void kernel_launch(void* const* d_in, const int* in_sizes, int n_in,
                              void* d_out, int out_size, void* d_ws, size_t ws_size,
                              hipStream_t stream) {
    const float* x     = (const float*)d_in[0];
    const int*   ei    = (const int*)d_in[1];
    const float* eattr = (const float*)d_in[2];
    const float* W1  = (const float*)d_in[3];  const float* We1 = (const float*)d_in[4];
    const float* as1 = (const float*)d_in[5];  const float* ad1 = (const float*)d_in[6];
    const float* ae1 = (const float*)d_in[7];  const float* b1  = (const float*)d_in[8];
    const float* W2  = (const float*)d_in[9];  const float* We2 = (const float*)d_in[10];
    const float* as2 = (const float*)d_in[11]; const float* ad2 = (const float*)d_in[12];
    const float* ae2 = (const float*)d_in[13]; const float* b2  = (const float*)d_in[14];
    const float* W3  = (const float*)d_in[15]; const float* We3 = (const float*)d_in[16];
    const float* as3 = (const float*)d_in[17]; const float* ad3 = (const float*)d_in[18];
    const float* ae3 = (const float*)d_in[19]; const float* b3  = (const float*)d_in[20];

    const int N = in_sizes[0] / 128;   // 50000
    const int E = in_sizes[1] / 2;     // 800000
    const int* srcp = ei;
    const int* dstp = ei + E;

    // ---- carve workspace ----
    char* ws = (char*)d_ws;
    size_t cur = 0;
    auto carve = [&](size_t bytes) -> void* {
        void* p = ws + cur;
        cur = (cur + bytes + 255) & ~(size_t)255;
        return p;
    };
    float*    loop_sum  = (float*)carve((size_t)N * 3 * sizeof(float));
    float*    loop_cnt  = (float*)carve((size_t)N * sizeof(float));
    float*    loop_attr = (float*)carve((size_t)N * 3 * sizeof(float));
    float*    we_att    = (float*)carve(16 * sizeof(float));
    float*    xl        = (float*)carve((size_t)N * 256 * sizeof(float));
    float*    h1        = (float*)carve((size_t)N * 256 * sizeof(float));
    float*    h2        = (float*)carve((size_t)N * 64 * sizeof(float));
    float*    a_s       = (float*)carve((size_t)N * 4 * sizeof(float));
    float*    a_d       = (float*)carve((size_t)N * 4 * sizeof(float));
    unsigned* mkey      = (unsigned*)carve((size_t)N * 4 * sizeof(unsigned));
    float*    denom     = (float*)carve((size_t)N * 4 * sizeof(float));
    float*    exbuf     = (float*)carve((size_t)(E + N) * 4 * sizeof(float));

    const int TPB = 256;

    // ---- self-loop attrs (fill_value='mean') ----
    k_zero<<<ugrid((long long)N * 3, TPB), TPB, 0, stream>>>(loop_sum, (long long)N * 3);
    k_zero<<<ugrid(N, TPB), TPB, 0, stream>>>(loop_cnt, N);
    k_loop_acc<<<ugrid(E, TPB), TPB, 0, stream>>>(dstp, eattr, loop_sum, loop_cnt, E);
    k_loop_fin<<<ugrid(N, TPB), TPB, 0, stream>>>(loop_sum, loop_cnt, loop_attr, N);

    auto run_layer = [&](const float* xin, int Din, int H, int C,
                         const float* W, const float* We, const float* as_,
                         const float* ad_, const float* ae_, const float* bb,
                         float* out, int do_elu) {
        const int HC = H * C;
        // fold We through att_e: 3 x H
        k_we_att<<<1, 64, 0, stream>>>(We, ae_, we_att, H, C);
        // xl = xin @ W   (fp32 WMMA)
        {
            long long mtiles = (N + 15) / 16, ntiles = HC / 16;
            long long tiles = mtiles * ntiles;
            int wavesPerBlock = 8;                       // 256 threads = 8 wave32
            long long blocks = (tiles + wavesPerBlock - 1) / wavesPerBlock;
            k_gemm_wmma_f32<<<(unsigned)blocks, wavesPerBlock * 32, 0, stream>>>(
                xin, W, xl, N, HC, Din);
        }
        // a_s, a_d
        k_node_att<<<ugrid((long long)N * H, TPB), TPB, 0, stream>>>(
            xl, as_, ad_, a_s, a_d, N, H, C);
        // zero accumulators (mkey zero == key(-inf))
        k_zero<<<ugrid((long long)N * H, TPB), TPB, 0, stream>>>((float*)mkey, (long long)N * H);
        k_zero<<<ugrid((long long)N * H, TPB), TPB, 0, stream>>>(denom, (long long)N * H);
        k_zero<<<ugrid((long long)N * HC, TPB), TPB, 0, stream>>>(out, (long long)N * HC);
        // edge passes
        long long tl = (long long)(E + N) * H;
        k_edge_logit<<<ugrid(tl, TPB), TPB, 0, stream>>>(
            srcp, dstp, eattr, loop_attr, we_att, a_s, a_d, exbuf, mkey, E, N, H);
        k_edge_exp<<<ugrid(tl, TPB), TPB, 0, stream>>>(dstp, exbuf, mkey, denom, E, N, H);
        long long ts = (long long)(E + N) * HC;
        k_edge_scatter<<<ugrid(ts, TPB), TPB, 0, stream>>>(
            srcp, dstp, exbuf, denom, xl, out, E, N, H, C);
        // bias (+ELU)
        k_bias_elu<<<ugrid((long long)N * HC, TPB), TPB, 0, stream>>>(
            out, bb, (long long)N * HC, HC, do_elu);
    };

    run_layer(x,  128, 4, 64, W1, We1, as1, ad1, ae1, b1, h1, 1);
    run_layer(h1, 256, 2, 32, W2, We2, as2, ad2, ae2, b2, h2, 1);
    run_layer(h2,  64, 1, 16, W3, We3, as3, ad3, ae3, b3, (float*)d_out, 0);
}